// ROCKET_21990232555994
// MI455X (gfx1250) — compile-verified
//
#include <hip/hip_runtime.h>
#include <hip/hip_bf16.h>

#define KNUM   10000
#define KLEN   11
#define LIN    512
#define BATCH  16
#define XSTR   520     // LDS row stride (floats): 4 zero pad | 512 data | 4 zero pad -> 16B-aligned data
#define NGROUPS 1024   // group id = min(dil,511)*2 + pad
#define MAXTILES 1664  // >= 10000/16 + NGROUPS

typedef __attribute__((ext_vector_type(2))) float v2f;
typedef __attribute__((ext_vector_type(8))) float v8f;

// workspace layout (int words)
#define WS_NUMTILES 0
#define WS_COUNTS   64
#define WS_OFFSETS  (WS_COUNTS + NGROUPS)
#define WS_CURSOR   (WS_OFFSETS + NGROUPS)
#define WS_PERM     (WS_CURSOR + NGROUPS)
#define WS_TSTART   (WS_PERM + KNUM)
#define WS_TCNT     (WS_TSTART + MAXTILES)
#define WS_TDIL     (WS_TCNT + MAXTILES)
#define WS_TPAD     (WS_TDIL + MAXTILES)

__device__ __forceinline__ int group_id(int dil, int pad) {
    dil = dil < 1 ? 1 : (dil > 511 ? 511 : dil);
    return dil * 2 + (pad & 1);
}
// reference clips gather index to [0, 513]; both ends map to zero sentinels in LDS
__device__ __forceinline__ int clamp_idx(int v) {
    return v < 0 ? 0 : (v > 513 ? 513 : v);
}

#if defined(__HIP_DEVICE_COMPILE__) && __has_builtin(__builtin_amdgcn_global_load_async_to_lds_b128)
#define USE_ASYNC_LDS 1
typedef int v4i __attribute__((vector_size(16)));
typedef __attribute__((address_space(1))) v4i v4i_g;   // global int4
typedef __attribute__((address_space(3))) v4i v4i_l;   // LDS int4
__device__ __forceinline__ v4i_g* to_glb_v4(const void* p) {
    return (v4i_g*)(unsigned long long)p;
}
__device__ __forceinline__ v4i_l* to_lds_v4(const void* p) {
    // generic LDS pointer: low 32 bits are the LDS byte offset (aperture in high bits)
    return (v4i_l*)(unsigned int)(unsigned long long)p;
}
#else
#define USE_ASYNC_LDS 0
#endif

__global__ __launch_bounds__(NGROUPS)
void rocket_zero(int* ws) { ws[WS_COUNTS + threadIdx.x] = 0; }

__global__ void rocket_count(const int* __restrict__ dil, const int* __restrict__ pad, int* ws) {
    int k = blockIdx.x * blockDim.x + threadIdx.x;
    if (k < KNUM) atomicAdd(&ws[WS_COUNTS + group_id(dil[k], pad[k])], 1);
}

__global__ __launch_bounds__(NGROUPS)
void rocket_scan(int* ws) {
    __shared__ int s[NGROUPS];
    int t = threadIdx.x;
    int c0 = ws[WS_COUNTS + t];
    s[t] = c0;
    __syncthreads();
    for (int off = 1; off < NGROUPS; off <<= 1) {
        int v = (t >= off) ? s[t - off] : 0;
        __syncthreads();
        s[t] += v;
        __syncthreads();
    }
    int excl = s[t] - c0;                 // exclusive prefix
    ws[WS_OFFSETS + t] = excl;
    ws[WS_CURSOR + t]  = excl;
    __syncthreads();
    if (t == 0) {
        int ti = 0;
        for (int g = 0; g < NGROUPS; ++g) {
            int c   = ws[WS_COUNTS + g];
            int off = ws[WS_OFFSETS + g];
            for (int i = 0; i < c; i += 16) {
                ws[WS_TSTART + ti] = off + i;
                ws[WS_TCNT + ti]   = (c - i < 16) ? (c - i) : 16;
                ws[WS_TDIL + ti]   = g >> 1;
                ws[WS_TPAD + ti]   = g & 1;
                ++ti;
            }
        }
        ws[WS_NUMTILES] = ti;
    }
}

__global__ void rocket_scatter(const int* __restrict__ dil, const int* __restrict__ pad, int* ws) {
    int k = blockIdx.x * blockDim.x + threadIdx.x;
    if (k < KNUM) {
        int g = group_id(dil[k], pad[k]);
        int pos = atomicAdd(&ws[WS_CURSOR + g], 1);
        ws[WS_PERM + pos] = k;
    }
}

// gather the three 16x4 A operands (f32) for one t-tile from LDS
__device__ __forceinline__ void loadA(const float* __restrict__ xb, int base, int dil, int h,
                                      v2f& a0, v2f& a1, v2f& a2) {
    int j = 2 * h;   // A VGPR r, lane half h -> tap K = j0 + r + 2h
    a0.x = xb[clamp_idx(base + (j + 0) * dil)];
    a0.y = xb[clamp_idx(base + (j + 1) * dil)];
    a1.x = xb[clamp_idx(base + (j + 4) * dil)];
    a1.y = xb[clamp_idx(base + (j + 5) * dil)];
    a2.x = xb[clamp_idx(base + (j + 8) * dil)];
    a2.y = xb[clamp_idx(base + (j + 9) * dil)];
}

// One block = one tile of <=16 kernels sharing (dilation, padding).
// 512 threads = 16 waves; wave w handles batch b = w.
// GEMM: D(16 t x 16 k) += A(16 t x 4 taps) * B(4 taps x 16 k), 3 chunks of V_WMMA_F32_16X16X4_F32.
__global__ __launch_bounds__(512)
void rocket_main(const float* __restrict__ X, const float* __restrict__ W,
                 const float* __restrict__ Bias, const int* __restrict__ Len,
                 float* __restrict__ out, const int* __restrict__ ws) {
    __shared__ float xs[BATCH * XSTR];   // 33.3 KB: [0..3]=0, [4..515]=x, [516..519]=0 per row
    __shared__ float wt[16 * 12];        // per-tile weights, tap 11 = 0
    __shared__ float bs[16];
    __shared__ int   lo[16];             // l_out per kernel
    __shared__ int   ki[16];             // original kernel index

    int nt = ws[WS_NUMTILES];
    if ((int)blockIdx.x >= nt) return;

    int tstart = ws[WS_TSTART + blockIdx.x];
    int tcnt   = ws[WS_TCNT + blockIdx.x];
    int dil    = ws[WS_TDIL + blockIdx.x];
    int pad    = ws[WS_TPAD + blockIdx.x];
    int start  = 1 - pad;

    int tid = threadIdx.x;

#if USE_ASYNC_LDS
    // 2048 16-byte chunks: chunk q -> b = q>>7, float col c = (q&127)*4; LDS dst 16B-aligned
    for (int q = tid; q < BATCH * 128; q += 512) {
        int b = q >> 7, c = (q & 127) << 2;
        __builtin_amdgcn_global_load_async_to_lds_b128(
            to_glb_v4(X + b * LIN + c), to_lds_v4(&xs[b * XSTR + 4 + c]), 0, 0);
    }
    // zero sentinels: cols 0..3 and 516..519
    for (int q = tid; q < BATCH * 8; q += 512) {
        int b = q >> 3, e = q & 7;
        xs[b * XSTR + (e < 4 ? e : 512 + e)] = 0.0f;
    }
#else
    for (int idx = tid; idx < BATCH * XSTR; idx += 512) {
        int b = idx / XSTR, c = idx - b * XSTR;
        float v = 0.0f;
        if (c >= 4 && c <= 515) v = X[b * LIN + (c - 4)];
        xs[idx] = v;
    }
#endif

    // stage per-kernel tile data (overlaps with async copy)
    if (tid < 16) {
        int n = tid;
        float wr[12];
        #pragma unroll
        for (int j = 0; j < 12; ++j) wr[j] = 0.0f;
        float bsv = 0.0f; int lov = 0; int kk = -1;
        if (n < tcnt) {
            kk = ws[WS_PERM + tstart + n];
            for (int j = 0; j < KLEN; ++j) wr[j] = W[kk * KLEN + j];
            bsv = Bias[kk];
            int len = Len[kk];
            lov = LIN + 2 * pad - dil * (len - 1);   // l_out (always <= 508)
        }
        #pragma unroll
        for (int j = 0; j < 12; ++j) wt[n * 12 + j] = wr[j];
        bs[n] = bsv; lo[n] = lov; ki[n] = kk;
    }

#if USE_ASYNC_LDS
#if __has_builtin(__builtin_amdgcn_s_wait_asynccnt)
    __builtin_amdgcn_s_wait_asynccnt(0);
#else
    asm volatile("s_wait_asynccnt 0x0" ::: "memory");
#endif
#endif
    __syncthreads();

    int lane = tid & 31;
    int wave = tid >> 5;           // batch index
    int h = lane >> 4;             // lane half: K offset 2h (A and B), M offset 8h (D)
    int m = lane & 15;             // A row t-offset / B kernel column n
    const float* xb = xs + wave * XSTR + 3;   // xb[i] for clipped i in [0,513]; i==0 / i==513 hit zeros

    // B operands (invariant over t-tiles): B VGPR r, half h -> tap j0 + r + 2h
    v2f b0, b1, b2;
    b0.x = wt[m * 12 + 0 + 2 * h]; b0.y = wt[m * 12 + 1 + 2 * h];
    b1.x = wt[m * 12 + 4 + 2 * h]; b1.y = wt[m * 12 + 5 + 2 * h];
    b2.x = wt[m * 12 + 8 + 2 * h]; b2.y = wt[m * 12 + 9 + 2 * h];
    float biasn = bs[m];
    int   lon   = lo[m];

    int   icnt = 0;
    float mx   = -__builtin_huge_valf();

    // software-pipelined: prefetch next tile's A gathers while current tile computes
    v2f a0, a1, a2;
    loadA(xb, start + m, dil, h, a0, a1, a2);

    for (int tt = 0; tt < 32; ++tt) {        // 32 tiles of 16 t's cover t=0..511 (masked to l_out<=508)
        int t0 = tt * 16;
        v2f p0 = a0, p1 = a1, p2 = a2;
        if (tt < 31) loadA(xb, start + t0 + 16 + m, dil, h, a0, a1, a2);

        v8f c;
        #pragma unroll
        for (int v = 0; v < 8; ++v) c[v] = biasn;
        c = __builtin_amdgcn_wmma_f32_16x16x4_f32(false, p0, false, b0, (short)0, c, false, false);
        c = __builtin_amdgcn_wmma_f32_16x16x4_f32(false, p1, false, b1, (short)0, c, false, false);
        c = __builtin_amdgcn_wmma_f32_16x16x4_f32(false, p2, false, b2, (short)0, c, false, false);

        #pragma unroll
        for (int v = 0; v < 8; ++v) {
            int t = t0 + v + 8 * h;            // D row M = v + 8h
            if (t < lon) {
                icnt += (c[v] > 0.0f) ? 1 : 0;
                mx = fmaxf(mx, c[v]);
            }
        }
    }

    // combine the two lane-halves (same kernel column n = lane&15)
    int   icnt2 = icnt + __shfl_xor(icnt, 16, 32);
    float mx2   = fmaxf(mx, __shfl_xor(mx, 16, 32));

    if (lane < 16 && lane < tcnt) {
        int kk = ki[lane];
        float denom = (float)lo[lane];
        out[wave * (2 * KNUM) + kk]        = (float)icnt2 / denom;   // ppv
        out[wave * (2 * KNUM) + KNUM + kk] = mx2;                    // max
    }
}

extern "C" void kernel_launch(void* const* d_in, const int* in_sizes, int n_in,
                              void* d_out, int out_size, void* d_ws, size_t ws_size,
                              hipStream_t stream) {
    (void)in_sizes; (void)n_in; (void)out_size; (void)ws_size;
    const float* X    = (const float*)d_in[0];
    const float* W    = (const float*)d_in[1];
    const float* Bias = (const float*)d_in[2];
    const int*   Len  = (const int*)d_in[3];
    const int*   Dil  = (const int*)d_in[4];
    const int*   Pad  = (const int*)d_in[5];
    float* out = (float*)d_out;
    int*   ws  = (int*)d_ws;

    rocket_zero   <<<1, NGROUPS, 0, stream>>>(ws);
    rocket_count  <<<(KNUM + 255) / 256, 256, 0, stream>>>(Dil, Pad, ws);
    rocket_scan   <<<1, NGROUPS, 0, stream>>>(ws);
    rocket_scatter<<<(KNUM + 255) / 256, 256, 0, stream>>>(Dil, Pad, ws);
    rocket_main   <<<MAXTILES, 512, 0, stream>>>(X, W, Bias, Len, out, ws);
}